// GraphPatchEmbed_18176301597543
// MI455X (gfx1250) — compile-verified
//
#include <hip/hip_runtime.h>
#include <hip/hip_bf16.h>

typedef __attribute__((ext_vector_type(2))) float v2f;
typedef __attribute__((ext_vector_type(8))) float v8f;

#define EMBED 192
#define KDIM  12      // 3 in-channels * 2*2 patch
#define HWN   65536   // 256*256 nodes per image
#define WGRID 256     // patch grid width

// ---------------------------------------------------------------------------
// Kernel 1: W_comb[k][c2] = sum_c conv_w[c*12+k] * gcn_w[c*192+c2]   (12x192)
// ---------------------------------------------------------------------------
__global__ void combine_weights_kernel(const float* __restrict__ conv_w,
                                       const float* __restrict__ gcn_w,
                                       float* __restrict__ w_comb) {
    int gid = blockIdx.x * blockDim.x + threadIdx.x;   // 0 .. 12*192-1
    if (gid >= KDIM * EMBED) return;
    int c2 = gid % EMBED;
    int k  = gid / EMBED;
    float s = 0.f;
    #pragma unroll 4
    for (int c = 0; c < EMBED; ++c)
        s += conv_w[c * KDIM + k] * gcn_w[c * EMBED + c2];
    w_comb[k * EMBED + c2] = s;
}

// ---------------------------------------------------------------------------
// Kernel 2: xw[node][c2] = P[node][0..11] @ W_comb  via V_WMMA_F32_16X16X4_F32
// Block: 256 threads (8 waves), 128 nodes x 192 channels per block.
// ---------------------------------------------------------------------------
__global__ void __launch_bounds__(256)
patch_gemm_kernel(const float* __restrict__ x,
                  const float* __restrict__ w_comb,
                  float* __restrict__ xw) {
    __shared__ float lds_W[KDIM * EMBED];   // [k][c2], stride 192
    __shared__ float lds_P[128 * KDIM];     // [node][k], stride 12 (conflict-free)

    const int tid  = threadIdx.x;
    const int wave = tid >> 5;
    const int lane = tid & 31;

    const int nodeBase = blockIdx.x * 128;
    const int b    = nodeBase >> 16;          // batch
    const int loc0 = nodeBase & (HWN - 1);
    const int h    = loc0 >> 8;               // patch row (constant in block)
    const int w0   = loc0 & (WGRID - 1);      // 0 or 128

    // Stage W_comb (2304 floats), coalesced
    #pragma unroll
    for (int it = 0; it < 9; ++it)
        lds_W[tid + it * 256] = w_comb[tid + it * 256];

    // Stage patch matrix P: 6 segments of 256 consecutive floats of x
    #pragma unroll
    for (int seg = 0; seg < 6; ++seg) {
        const int i  = seg >> 1;              // in-channel
        const int kh = seg & 1;
        const int row = 2 * h + kh;
        const float v = x[(((size_t)(b * 3 + i) * 512) + row) * 512 + 2 * w0 + tid];
        lds_P[(tid >> 1) * KDIM + (i * 4 + kh * 2 + (tid & 1))] = v;
    }
    __syncthreads();

    // A fragments: ISA 32-bit 16x4 layout. lanes0-15: K=kb+0/kb+1; lanes16-31: K=kb+2/kb+3
    const int m    = lane & 15;
    const int khalf = (lane < 16) ? 0 : 2;
    v2f a[3];
    #pragma unroll
    for (int kc = 0; kc < 3; ++kc) {
        const float* pr = &lds_P[(wave * 16 + m) * KDIM + kc * 4 + khalf];
        a[kc].x = pr[0];
        a[kc].y = pr[1];
    }

    // 12 N-tiles of 16 channels each
    #pragma unroll 4
    for (int j = 0; j < 12; ++j) {
        v8f acc = {};
        #pragma unroll
        for (int kc = 0; kc < 3; ++kc) {
            v2f bf;
            bf.x = lds_W[(kc * 4 + khalf + 0) * EMBED + j * 16 + m];
            bf.y = lds_W[(kc * 4 + khalf + 1) * EMBED + j * 16 + m];
            acc = __builtin_amdgcn_wmma_f32_16x16x4_f32(
                false, a[kc], false, bf, (short)0, acc, false, false);
        }
        // D layout: VGPR r -> lanes0-15 (M=r), lanes16-31 (M=r+8)
        const int mAdd = (lane < 16) ? 0 : 8;
        const int ch   = j * 16 + m;
        #pragma unroll
        for (int r = 0; r < 8; ++r) {
            const int node = nodeBase + wave * 16 + r + mAdd;
            xw[(size_t)node * EMBED + ch] = acc[r];
        }
    }
}

// ---------------------------------------------------------------------------
// Kernel 3: normalized aggregation + bias. One block per node, 192 threads.
// ---------------------------------------------------------------------------
__device__ __forceinline__ float deg_of(int h, int w) {
    int nb = (h > 0) + (h < 255) + (w > 0) + (w < 255);
    float d = (float)(1 + nb);
    if (h == 254 && w == 254) d += 1.0f;   // buggy extra edge target
    return d;
}

__global__ void __launch_bounds__(192)
gcn_aggregate_kernel(const float* __restrict__ xw,
                     const float* __restrict__ gcn_b,
                     float* __restrict__ out) {
    const int node = blockIdx.x;            // 0 .. 262143
    const int c    = threadIdx.x;           // 0 .. 191
    const float bias = gcn_b[c];
    const size_t self = (size_t)node * EMBED + c;

    const int b = node >> 16;
    if (b > 0) {                            // self-loop only: deg=1, norm=1
        out[self] = xw[self] + bias;
        return;
    }
    const int loc = node & (HWN - 1);
    const int h = loc >> 8;
    const int w = loc & (WGRID - 1);

    const float dself = rsqrtf(deg_of(h, w));
    float s = xw[self] * dself * dself;
    if (h > 0)   s += xw[self - (size_t)WGRID * EMBED] * rsqrtf(deg_of(h - 1, w)) * dself;
    if (h < 255) s += xw[self + (size_t)WGRID * EMBED] * rsqrtf(deg_of(h + 1, w)) * dself;
    if (w > 0)   s += xw[self - EMBED]                 * rsqrtf(deg_of(h, w - 1)) * dself;
    if (w < 255) s += xw[self + EMBED]                 * rsqrtf(deg_of(h, w + 1)) * dself;
    if (loc == HWN - 1 - WGRID - 1)        // node (254,254): msg from (255,255), corner deg=3
        s += xw[(size_t)(HWN - 1) * EMBED + c] * rsqrtf(3.0f) * dself;
    out[self] = s + bias;
}

// ---------------------------------------------------------------------------
extern "C" void kernel_launch(void* const* d_in, const int* in_sizes, int n_in,
                              void* d_out, int out_size, void* d_ws, size_t ws_size,
                              hipStream_t stream) {
    const float* x      = (const float*)d_in[0];  // (4,3,512,512)
    const float* conv_w = (const float*)d_in[1];  // (192,3,2,2)
    const float* gcn_w  = (const float*)d_in[2];  // (192,192)
    const float* gcn_b  = (const float*)d_in[3];  // (192,)
    float* out = (float*)d_out;                   // (4,65536,192)

    float* w_comb = (float*)d_ws;                 // 12*192 floats
    float* xw     = (float*)d_ws + 4096;          // 16KB offset; 4*65536*192 floats

    // 1) combine conv + gcn weights into 12x192
    combine_weights_kernel<<<9, 256, 0, stream>>>(conv_w, gcn_w, w_comb);

    // 2) fused patch-embed + linear via fp32 WMMA: 262144 nodes / 128 per block
    patch_gemm_kernel<<<2048, 256, 0, stream>>>(x, w_comb, xw);

    // 3) normalized aggregation + bias
    gcn_aggregate_kernel<<<262144, 192, 0, stream>>>(xw, gcn_b, out);
}